// WaveConv2dCwt_4535485464999
// MI455X (gfx1250) — compile-verified
//
#include <hip/hip_runtime.h>

typedef __attribute__((ext_vector_type(2))) float v2f;
typedef __attribute__((ext_vector_type(8))) float v8f;

#define RSQRT2 0.70710678118654752440f

// Filter table (derived from reference):
//  H0A = reverse(H0B); H1A[k] = (-1)^k H0B[k]; H1B = reverse(H1A)
//  G0A,G0B,G1A,G1B = H0B,H0A,H1B,H1A
//  colifilt phase filters: even/odd output taps of the dilated filters.
__device__ __constant__ float FILT[72] = {
  /* 0  H0O(5)  */ -0.05f, 0.25f, 0.6f, 0.25f, -0.05f,
  /* 5  G0O(7)  */ -0.010714285714f, -0.053571428571f, 0.260714285714f,
                    0.607142857143f,  0.260714285714f, -0.053571428571f, -0.010714285714f,
  /* 12 H0B(10) */ 0.03516384f, 0.0f, -0.08832942f, 0.23389032f, 0.76027237f,
                   0.5875183f, 0.0f, -0.11430184f, 0.0f, 0.0f,
  /* 22 H0A(10) */ 0.0f, 0.0f, -0.11430184f, 0.0f, 0.5875183f,
                   0.76027237f, 0.23389032f, -0.08832942f, 0.0f, 0.03516384f,
  /* 32 H1A(10) */ 0.03516384f, 0.0f, -0.08832942f, -0.23389032f, 0.76027237f,
                   -0.5875183f, 0.0f, 0.11430184f, 0.0f, 0.0f,
  /* 42 H1B(10) */ 0.0f, 0.0f, 0.11430184f, 0.0f, -0.5875183f,
                   0.76027237f, -0.23389032f, -0.08832942f, 0.0f, 0.03516384f,
  /* 52 G0 even phase (src off +1): G0B=H0A odd taps  */
                   0.0f, 0.0f, 0.76027237f, -0.08832942f, 0.03516384f,
  /* 57 G0 odd  phase (src off +2): G0A=H0B even taps */
                   0.03516384f, -0.08832942f, 0.76027237f, 0.0f, 0.0f,
  /* 62 G1 even phase (src off +2): G1A=H1B even taps */
                   0.0f, 0.11430184f, -0.5875183f, -0.23389032f, 0.0f,
  /* 67 G1 odd  phase (src off +1): G1B=H1A odd taps  */
                   0.0f, -0.23389032f, -0.5875183f, 0.11430184f, 0.0f
};
enum { F_H0O = 0, F_G0O = 5, F_H0B = 12, F_H0A = 22, F_H1A = 32, F_H1B = 42,
       F_G0E = 52, F_G0D = 57, F_G1E = 62, F_G1D = 67 };

__device__ __forceinline__ int refl(int p, int L) {
  p = (p < 0) ? (-1 - p) : p;
  p = (p >= L) ? (2 * L - 1 - p) : p;
  return p;
}

// ---------------- stride-1 symmetric filters (colfilter) ----------------
// y(n,i,j) = sum_t h[t] * x(n, refl(i+m2-t), j)
__global__ void k_colfilter_v(const float* __restrict__ x, float* __restrict__ y,
                              int N, int H, int W, int f, int taps, int m2) {
  int idx = blockIdx.x * blockDim.x + threadIdx.x;
  if (idx >= N * H * W) return;
  int j = idx % W;
  int i = (idx / W) % H;
  const float* xp = x + (idx / (H * W)) * H * W;
  float acc = 0.f;
  for (int t = 0; t < taps; ++t)
    acc += FILT[f + t] * xp[refl(i + m2 - t, H) * W + j];
  y[idx] = acc;
}

__global__ void k_colfilter_h(const float* __restrict__ x, float* __restrict__ y,
                              int N, int H, int W, int f, int taps, int m2) {
  int idx = blockIdx.x * blockDim.x + threadIdx.x;
  if (idx >= N * H * W) return;
  int j = idx % W;
  int i = (idx / W) % H;
  const float* xp = x + (idx / (H * W)) * H * W + i * W;
  float acc = 0.f;
  for (int t = 0; t < taps; ++t)
    acc += FILT[f + t] * xp[refl(j + m2 - t, W)];
  y[idx] = acc;
}

// ---------------- coldfilt: decimate by 2 (q-shift, len-10) ----------------
// out row 2u   = sum_t fe[t] x_sym[4u+4-t]
// out row 2u+1 = sum_t fo[t] x_sym[4u+4-t]
__global__ void k_coldfilt_v(const float* __restrict__ x, float* __restrict__ y,
                             int N, int H, int W, int fe, int fo) {
  int Ho = H >> 1;
  int idx = blockIdx.x * blockDim.x + threadIdx.x;
  if (idx >= N * Ho * W) return;
  int j = idx % W;
  int io = (idx / W) % Ho;
  const float* xp = x + (idx / (Ho * W)) * H * W;
  int f = (io & 1) ? fo : fe;
  int base = 4 * (io >> 1) + 4;
  float acc = 0.f;
#pragma unroll
  for (int t = 0; t < 10; ++t)
    acc += FILT[f + t] * xp[refl(base - t, H) * W + j];
  y[idx] = acc;
}

__global__ void k_coldfilt_h(const float* __restrict__ x, float* __restrict__ y,
                             int N, int H, int W, int fe, int fo) {
  int Wo = W >> 1;
  int idx = blockIdx.x * blockDim.x + threadIdx.x;
  if (idx >= N * H * Wo) return;
  int jo = idx % Wo;
  int i = (idx / Wo) % H;
  const float* xp = x + (idx / (H * Wo)) * H * W + i * W;
  int f = (jo & 1) ? fo : fe;
  int base = 4 * (jo >> 1) + 4;
  float acc = 0.f;
#pragma unroll
  for (int t = 0; t < 10; ++t)
    acc += FILT[f + t] * xp[refl(base - t, W)];
  y[idx] = acc;
}

// ---------------- colifilt: upsample by 2, dual-input fused add ----------------
// x0 filtered with G0 pair, x1 (nullable) with G1 pair
__global__ void k_colifilt_v_dual(const float* __restrict__ x0, const float* __restrict__ x1,
                                  float* __restrict__ y, int N, int H, int W) {
  int Ho = H << 1;
  int idx = blockIdx.x * blockDim.x + threadIdx.x;
  if (idx >= N * Ho * W) return;
  int j = idx % W;
  int io = (idx / W) % Ho;
  int n = idx / (Ho * W);
  int i = io >> 1;
  float acc = 0.f;
  const float* p0 = x0 + n * H * W;
  if ((io & 1) == 0) {
#pragma unroll
    for (int r = 0; r < 5; ++r) acc += FILT[F_G0E + r] * p0[refl(i + 1 - r, H) * W + j];
    if (x1) {
      const float* p1 = x1 + n * H * W;
#pragma unroll
      for (int r = 0; r < 5; ++r) acc += FILT[F_G1E + r] * p1[refl(i + 2 - r, H) * W + j];
    }
  } else {
#pragma unroll
    for (int r = 0; r < 5; ++r) acc += FILT[F_G0D + r] * p0[refl(i + 2 - r, H) * W + j];
    if (x1) {
      const float* p1 = x1 + n * H * W;
#pragma unroll
      for (int r = 0; r < 5; ++r) acc += FILT[F_G1D + r] * p1[refl(i + 1 - r, H) * W + j];
    }
  }
  y[idx] = acc;
}

__global__ void k_colifilt_h_dual(const float* __restrict__ x0, const float* __restrict__ x1,
                                  float* __restrict__ y, int N, int H, int W) {
  int Wo = W << 1;
  int idx = blockIdx.x * blockDim.x + threadIdx.x;
  if (idx >= N * H * Wo) return;
  int jo = idx % Wo;
  int i = (idx / Wo) % H;
  int n = idx / (H * Wo);
  int j = jo >> 1;
  float acc = 0.f;
  const float* p0 = x0 + n * H * W + i * W;
  if ((jo & 1) == 0) {
#pragma unroll
    for (int r = 0; r < 5; ++r) acc += FILT[F_G0E + r] * p0[refl(j + 1 - r, W)];
    if (x1) {
      const float* p1 = x1 + n * H * W + i * W;
#pragma unroll
      for (int r = 0; r < 5; ++r) acc += FILT[F_G1E + r] * p1[refl(j + 2 - r, W)];
    }
  } else {
#pragma unroll
    for (int r = 0; r < 5; ++r) acc += FILT[F_G0D + r] * p0[refl(j + 2 - r, W)];
    if (x1) {
      const float* p1 = x1 + n * H * W + i * W;
#pragma unroll
      for (int r = 0; r < 5; ++r) acc += FILT[F_G1D + r] * p1[refl(j + 1 - r, W)];
    }
  }
  y[idx] = acc;
}

// ---------------- q2c pack: bands (256,32,32) -> R/I coeffs (6,256,16,16) ----------------
__global__ void k_q2c_pack(const float* __restrict__ bh, const float* __restrict__ bv,
                           const float* __restrict__ bd,
                           float* __restrict__ R, float* __restrict__ I) {
  int idx = blockIdx.x * blockDim.x + threadIdx.x;
  if (idx >= 256 * 256) return;
  int n = idx >> 8;
  int xy = idx & 255;
  int x = xy >> 4, y = xy & 15;
  auto doband = [&](const float* band, int s1, int s2) {
    const float* p = band + n * 1024;
    float a = p[(2 * x) * 32 + 2 * y];
    float b = p[(2 * x) * 32 + 2 * y + 1];
    float c = p[(2 * x + 1) * 32 + 2 * y];
    float d = p[(2 * x + 1) * 32 + 2 * y + 1];
    R[(s1 * 256 + n) * 256 + xy] = (a - d) * RSQRT2;
    I[(s1 * 256 + n) * 256 + xy] = (b + c) * RSQRT2;
    R[(s2 * 256 + n) * 256 + xy] = (a + d) * RSQRT2;
    I[(s2 * 256 + n) * 256 + xy] = (b - c) * RSQRT2;
  };
  doband(bh, 0, 5);
  doband(bd, 1, 4);
  doband(bv, 2, 3);
}

// ---------------- c2q unpack: R/I (6,256,16,16) -> bands (256,32,32) ----------------
__global__ void k_c2q_unpack(const float* __restrict__ R, const float* __restrict__ I,
                             float* __restrict__ bh, float* __restrict__ bv,
                             float* __restrict__ bd) {
  int idx = blockIdx.x * blockDim.x + threadIdx.x;
  if (idx >= 256 * 256) return;
  int n = idx >> 8;
  int xy = idx & 255;
  int x = xy >> 4, y = xy & 15;
  auto doband = [&](float* band, int s1, int s2) {
    float z1r = R[(s1 * 256 + n) * 256 + xy];
    float z1i = I[(s1 * 256 + n) * 256 + xy];
    float z2r = R[(s2 * 256 + n) * 256 + xy];
    float z2i = I[(s2 * 256 + n) * 256 + xy];
    float* p = band + n * 1024;
    p[(2 * x) * 32 + 2 * y]         = (z1r + z2r) * RSQRT2;
    p[(2 * x) * 32 + 2 * y + 1]     = (z1i + z2i) * RSQRT2;
    p[(2 * x + 1) * 32 + 2 * y]     = (z1i - z2i) * RSQRT2;
    p[(2 * x + 1) * 32 + 2 * y + 1] = (z2r - z1r) * RSQRT2;
  };
  doband(bh, 0, 5);
  doband(bd, 1, 4);
  doband(bv, 2, 3);
}

// ---------------- WMMA einsum: out_ll(b,o,x,y) = sum_i ll4(b,i,x,y) w0(i,o,x,y) ----------------
// One wave per (position, o-tile). M=16 (rows 0..3 = batch), N=16 o, K=4 per WMMA.
// Branch-free: invalid M rows load row 0 and are multiplied by 0 (no EXEC churn,
// tight clause of 4 loads + 1 v_wmma per K-step).
__global__ __launch_bounds__(32) void k_einsum_ll(const float* __restrict__ ll4,
                                                 const float* __restrict__ w0,
                                                 float* __restrict__ out) {
  int wid = blockIdx.x;          // 1024 positions * 4 o-tiles
  int o0 = (wid & 3) * 16;
  int xy = wid >> 2;             // 0..1023
  int lane = threadIdx.x;
  int m = lane & 15;             // M (A) / N (B) index
  int kh = (lane >> 4) << 1;     // K half: 0 or 2
  float msk = (m < 4) ? 1.f : 0.f;
  int row = (m < 4) ? m : 0;
  const float* ap = ll4 + row * 64 * 1024 + kh * 1024 + xy;          // + k0*1024
  const float* bp = w0 + (kh * 64 + o0 + m) * 1024 + xy;             // + k0*64*1024
  v8f acc = {};
#pragma unroll
  for (int k0 = 0; k0 < 64; k0 += 4) {
    v2f a, b;
    a[0] = msk * ap[k0 * 1024];
    a[1] = msk * ap[k0 * 1024 + 1024];
    b[0] = bp[k0 * 64 * 1024];
    b[1] = bp[k0 * 64 * 1024 + 64 * 1024];
    acc = __builtin_amdgcn_wmma_f32_16x16x4_f32(false, a, false, b, (short)0, acc,
                                                false, false);
  }
  if (lane < 16) {
#pragma unroll
    for (int r = 0; r < 4; ++r)
      out[(r * 64 + o0 + lane) * 1024 + xy] = acc[r];
  }
}

// ---------------- WMMA band einsum: real (Wr) + imag (Wc) chains per wave ----------------
// Rc/Ic layout (6,4,64,16,16); Wr/Wc (6,64,64,16,16); outputs same layout as Rc/Ic.
__global__ __launch_bounds__(32) void k_einsum_band(const float* __restrict__ Rc,
                                                   const float* __restrict__ Ic,
                                                   const float* __restrict__ Wr,
                                                   const float* __restrict__ Wc,
                                                   float* __restrict__ oR,
                                                   float* __restrict__ oI) {
  int wid = blockIdx.x;          // 6 * 256 positions * 4 o-tiles
  int o0 = (wid & 3) * 16;
  int rest = wid >> 2;
  int xy = rest & 255;
  int s = rest >> 8;
  int lane = threadIdx.x;
  int m = lane & 15;
  int kh = (lane >> 4) << 1;
  float msk = (m < 4) ? 1.f : 0.f;
  int row = (m < 4) ? m : 0;
  int aoff = ((s * 4 + row) * 64 + kh) * 256 + xy;                   // + k0*256
  int boff = ((s * 64 + kh) * 64 + o0 + m) * 256 + xy;               // + k0*64*256
  const float* arp = Rc + aoff;
  const float* aip = Ic + aoff;
  const float* brp = Wr + boff;
  const float* bip = Wc + boff;
  v8f accR = {}, accI = {};
#pragma unroll
  for (int k0 = 0; k0 < 64; k0 += 4) {
    v2f ar, ai, br, bi;
    ar[0] = msk * arp[k0 * 256];
    ar[1] = msk * arp[k0 * 256 + 256];
    ai[0] = msk * aip[k0 * 256];
    ai[1] = msk * aip[k0 * 256 + 256];
    br[0] = brp[k0 * 64 * 256];
    br[1] = brp[k0 * 64 * 256 + 64 * 256];
    bi[0] = bip[k0 * 64 * 256];
    bi[1] = bip[k0 * 64 * 256 + 64 * 256];
    accR = __builtin_amdgcn_wmma_f32_16x16x4_f32(false, ar, false, br, (short)0, accR,
                                                 false, false);
    accI = __builtin_amdgcn_wmma_f32_16x16x4_f32(false, ai, false, bi, (short)0, accI,
                                                 false, false);
  }
  if (lane < 16) {
#pragma unroll
    for (int r = 0; r < 4; ++r) {
      oR[((s * 4 + r) * 64 + o0 + lane) * 256 + xy] = accR[r];
      oI[((s * 4 + r) * 64 + o0 + lane) * 256 + xy] = accI[r];
    }
  }
}

static inline int nb(int tot) { return (tot + 255) / 256; }

extern "C" void kernel_launch(void* const* d_in, const int* in_sizes, int n_in,
                              void* d_out, int out_size, void* d_ws, size_t ws_size,
                              hipStream_t stream) {
  (void)in_sizes; (void)n_in; (void)out_size; (void)ws_size;
  const float* x  = (const float*)d_in[0];   // (4,64,256,256)
  const float* w0 = (const float*)d_in[1];   // (64,64,32,32)
  const float* Wr = (const float*)d_in[2];   // (6,64,64,16,16)
  const float* Wc = (const float*)d_in[3];   // (6,64,64,16,16)
  float* out = (float*)d_out;                // (4,64,256,256)

  char* ws = (char*)d_ws;
  const size_t MB = (size_t)1 << 20;
  auto F = [&](size_t off) { return (float*)(ws + off); };

  const int N = 256;  // B*C

  // workspace layout (peak ~192 MB, heavy reuse)
  float* t1   = F(0);          // 64MB   (N,256,256)
  float* ll1  = F(64 * MB);    // 64MB   (N,256,256)
  float* lo2  = F(0);          // 32MB   (N,128,256)  reuses t1
  float* ll2  = F(128 * MB);   // 16MB   (N,128,128)
  float* lo3  = F(144 * MB);   //  8MB   (N,64,128)
  float* ll3  = F(152 * MB);   //  4MB   (N,64,64)
  float* lo4  = F(156 * MB);   //  2MB   (N,32,64)
  float* hi4  = F(158 * MB);   //  2MB   (N,32,64)
  float* ll4  = F(160 * MB);   //  1MB   (N,32,32)
  float* bh   = F(161 * MB);
  float* bv   = F(162 * MB);
  float* bd   = F(163 * MB);
  float* Rc   = F(164 * MB);   // 1.5MB  (6,256,16,16)
  float* Ic   = F(166 * MB);
  float* oll  = F(168 * MB);   //  1MB
  float* oR   = F(169 * MB);   // 1.5MB
  float* oI   = F(171 * MB);
  float* bh2  = F(173 * MB);
  float* bv2  = F(174 * MB);
  float* bd2  = F(175 * MB);
  float* loj3 = F(176 * MB);   //  2MB   (N,32,64)
  float* hij3 = F(178 * MB);   //  2MB
  float* ll3i = F(180 * MB);   //  4MB   (N,64,64)
  float* lo2i = F(184 * MB);   //  8MB   (N,64,128)
  float* ll2i = F(0);          // 16MB   (N,128,128)  reuses A
  float* lo1i = F(16 * MB);    // 32MB   (N,128,256)
  float* ll1i = F(64 * MB);    // 64MB   (N,256,256)  reuses ll1
  float* lo0  = F(0);          // 64MB   reuses A

  // ---------- forward (only the path that survives the zeroed subbands) ----------
  k_colfilter_v<<<nb(N * 256 * 256), 256, 0, stream>>>(x, t1, N, 256, 256, F_H0O, 5, 2);
  k_colfilter_h<<<nb(N * 256 * 256), 256, 0, stream>>>(t1, ll1, N, 256, 256, F_H0O, 5, 2);
  k_coldfilt_v<<<nb(N * 128 * 256), 256, 0, stream>>>(ll1, lo2, N, 256, 256, F_H0B, F_H0A);
  k_coldfilt_h<<<nb(N * 128 * 128), 256, 0, stream>>>(lo2, ll2, N, 128, 256, F_H0B, F_H0A);
  k_coldfilt_v<<<nb(N * 64 * 128), 256, 0, stream>>>(ll2, lo3, N, 128, 128, F_H0B, F_H0A);
  k_coldfilt_h<<<nb(N * 64 * 64), 256, 0, stream>>>(lo3, ll3, N, 64, 128, F_H0B, F_H0A);
  k_coldfilt_v<<<nb(N * 32 * 64), 256, 0, stream>>>(ll3, lo4, N, 64, 64, F_H0B, F_H0A);
  k_coldfilt_v<<<nb(N * 32 * 64), 256, 0, stream>>>(ll3, hi4, N, 64, 64, F_H1A, F_H1B);
  k_coldfilt_h<<<nb(N * 32 * 32), 256, 0, stream>>>(lo4, ll4, N, 32, 64, F_H0B, F_H0A);
  k_coldfilt_h<<<nb(N * 32 * 32), 256, 0, stream>>>(lo4, bh, N, 32, 64, F_H1A, F_H1B);
  k_coldfilt_h<<<nb(N * 32 * 32), 256, 0, stream>>>(hi4, bv, N, 32, 64, F_H0B, F_H0A);
  k_coldfilt_h<<<nb(N * 32 * 32), 256, 0, stream>>>(hi4, bd, N, 32, 64, F_H1A, F_H1B);
  k_q2c_pack<<<nb(256 * 256), 256, 0, stream>>>(bh, bv, bd, Rc, Ic);

  // ---------- WMMA channel-mixing einsums ----------
  k_einsum_ll<<<1024 * 4, 32, 0, stream>>>(ll4, w0, oll);
  k_einsum_band<<<6 * 256 * 4, 32, 0, stream>>>(Rc, Ic, Wr, Wc, oR, oI);
  k_c2q_unpack<<<nb(256 * 256), 256, 0, stream>>>(oR, oI, bh2, bv2, bd2);

  // ---------- inverse ----------
  // level 4 (with bands)
  k_colifilt_h_dual<<<nb(N * 32 * 64), 256, 0, stream>>>(oll, bh2, loj3, N, 32, 32);
  k_colifilt_h_dual<<<nb(N * 32 * 64), 256, 0, stream>>>(bv2, bd2, hij3, N, 32, 32);
  k_colifilt_v_dual<<<nb(N * 64 * 64), 256, 0, stream>>>(loj3, hij3, ll3i, N, 32, 64);
  // levels 3,2 (bands are zero -> lo-only chain)
  k_colifilt_h_dual<<<nb(N * 64 * 128), 256, 0, stream>>>(ll3i, nullptr, lo2i, N, 64, 64);
  k_colifilt_v_dual<<<nb(N * 128 * 128), 256, 0, stream>>>(lo2i, nullptr, ll2i, N, 64, 128);
  k_colifilt_h_dual<<<nb(N * 128 * 256), 256, 0, stream>>>(ll2i, nullptr, lo1i, N, 128, 128);
  k_colifilt_v_dual<<<nb(N * 256 * 256), 256, 0, stream>>>(lo1i, nullptr, ll1i, N, 128, 256);
  // level 1 biorthogonal synthesis (bands zero)
  k_colfilter_h<<<nb(N * 256 * 256), 256, 0, stream>>>(ll1i, lo0, N, 256, 256, F_G0O, 7, 3);
  k_colfilter_v<<<nb(N * 256 * 256), 256, 0, stream>>>(lo0, out, N, 256, 256, F_G0O, 7, 3);
}